// CartesianDensityBlock_27943057228070
// MI455X (gfx1250) — compile-verified
//
#include <hip/hip_runtime.h>
#include <math.h>

// ---------------------------------------------------------------------------
// CartesianDensityBlock for MI455X (gfx1250, wave32)
//
// Stage A: zero d_out (the den accumulators live in-place in d_out).
// Stage B: scatter-add msg0/1/2 * (1/sqrt(50)) into den0/1/2 with
//          global_atomic_add_f32 (entire 166MB target fits in 192MB L2;
//          edge stream read non-temporally so it doesn't evict the L2 accum).
// Stage C: one wave per 16-node tile does the whole per-node math with
//          V_WMMA_F32_16X16X4_F32 (f32 in / f32 acc == reference precision).
// ---------------------------------------------------------------------------

#define INV_SQRT_DEG 0.14142135623730950488f

typedef __attribute__((ext_vector_type(2))) float v2f;
typedef __attribute__((ext_vector_type(8))) float v8f;

__device__ __forceinline__ v8f wmma_f32_k4(v2f a, v2f b, v8f c) {
  // D(16x16,f32) = A(16x4,f32) * B(4x16,f32) + C
  return __builtin_amdgcn_wmma_f32_16x16x4_f32(
      /*neg_a=*/false, a, /*neg_b=*/false, b,
      /*c_mod=*/(short)0, c, /*reuse_a=*/false, /*reuse_b=*/false);
}

__device__ __forceinline__ float silu(float x) {
  return x / (1.0f + __expf(-x));
}

// X(16 rows x K) @ W^T(K x 16 cols starting at col0).
// A rows come from LDS (padded stride), B rows are weight rows in global
// (tiny, L2/L0 resident). Per ISA VGPR layouts:
//   A: lane l16 -> row, VGPR0/1 -> K = k+2*half, k+2*half+1
//   B: lane l16 -> col, VGPR0/1 -> same K pair
//   C: VGPR r -> row r+8*half, col l16
__device__ __forceinline__ v8f gemm_tile_lds(const float* As, int astride,
                                             const float* __restrict__ W,
                                             int wstride, int K, int col0,
                                             int l16, int half) {
  v8f c = {0.f, 0.f, 0.f, 0.f, 0.f, 0.f, 0.f, 0.f};
  const float* wrow = W + (size_t)(col0 + l16) * (size_t)wstride;
  const float* arow = As + (size_t)l16 * (size_t)astride;
  for (int k = 0; k < K; k += 4) {
    int kk = k + 2 * half;
    v2f a;
    a.x = arow[kk];
    a.y = arow[kk + 1];
    v2f b;
    b.x = wrow[kk];
    b.y = wrow[kk + 1];
    c = wmma_f32_k4(a, b, c);
  }
  return c;
}

// ---------------------------------------------------------------------------
// Stage A: zero the output/accumulator buffer (16B stores).
// ---------------------------------------------------------------------------
__global__ __launch_bounds__(256) void zero_f4_kernel(float4* __restrict__ p,
                                                      long n4) {
  long i = (long)blockIdx.x * blockDim.x + threadIdx.x;
  if (i < n4) p[i] = make_float4(0.f, 0.f, 0.f, 0.f);
}

// ---------------------------------------------------------------------------
// Stage B: scatter-add one msg tensor into its den region.
// PER = floats per edge (64 / 192 / 576). Consecutive threads share an edge
// (index load broadcasts; msg read is coalesced & non-temporal).
// ---------------------------------------------------------------------------
__device__ __forceinline__ void atomic_add_f32(float* p, float v) {
#if defined(__HIP_DEVICE_COMPILE__)
  unsafeAtomicAdd(p, v);  // -> global_atomic_add_f32 (no CAS loop)
#else
  atomicAdd(p, v);
#endif
}

template <int PER>
__global__ __launch_bounds__(256) void scatter_add_kernel(
    const float* __restrict__ msg, const int* __restrict__ index,
    float* __restrict__ den, long total) {
  long i = (long)blockIdx.x * blockDim.x + threadIdx.x;
  if (i >= total) return;
  int e = (int)(i / PER);                 // constant divisor -> mul/shift
  int s = (int)(i - (long)e * PER);
  int node = index[e];
  float v = __builtin_nontemporal_load(msg + i) * INV_SQRT_DEG;
  atomic_add_f32(den + (size_t)node * PER + s, v);
}

// ---------------------------------------------------------------------------
// Stage C: per-node math. One wave (32 lanes) per tile of 16 nodes.
// All matmuls via V_WMMA_F32_16X16X4_F32. In-place: reads den* from d_out,
// writes delta_h* back to the same locations.
// ---------------------------------------------------------------------------
__global__ __launch_bounds__(32) void node_update_kernel(
    float* __restrict__ out, int n_nodes,
    const float* __restrict__ W_s1, const float* __restrict__ b_s1,
    const float* __restrict__ W_s2, const float* __restrict__ b_s2,
    const float* __restrict__ W_g1, const float* __restrict__ b_g1,
    const float* __restrict__ W_g2, const float* __restrict__ b_g2,
    const float* __restrict__ W_L1, const float* __restrict__ W_L2) {
  // Odd-padded strides -> no 16-way LDS bank conflicts on A-fragment reads.
  __shared__ float s_concat[16][193];   // [den0 | inv1 | inv2], later [dh0 | g]
  __shared__ float s_x[16][65];         // h, then staged den1/den2 tiles
  __shared__ float s_scales[16][129];   // [alpha1 | alpha2]

  const int tid = threadIdx.x;
  const int lane = tid & 31;
  const int half = lane >> 4;
  const int l16 = lane & 15;
  const int base = blockIdx.x * 16;

  float* __restrict__ den0 = out;
  float* __restrict__ den1 = out + (size_t)n_nodes * 64;
  float* __restrict__ den2 = out + (size_t)n_nodes * 256;

  // ---- Phase 0: stage concat = [den0, inv1, inv2] into LDS --------------
  for (int t = tid; t < 16 * 64; t += 32) {
    int n = t >> 6, f = t & 63;
    int node = base + n;
    if (node >= n_nodes) node = n_nodes - 1;  // tail rows: garbage, never stored
    s_concat[n][f] = den0[(size_t)node * 64 + f];
    const float* p1 = den1 + (size_t)node * 192 + f;
    float s1 = 1e-8f;
#pragma unroll
    for (int i = 0; i < 3; ++i) { float v = p1[i * 64]; s1 += v * v; }
    s_concat[n][64 + f] = sqrtf(s1);
    const float* p2 = den2 + (size_t)node * 576 + f;
    float s2 = 1e-8f;
#pragma unroll
    for (int i = 0; i < 9; ++i) { float v = p2[i * 64]; s2 += v * v; }
    s_concat[n][128 + f] = sqrtf(s2);
  }
  __syncthreads();

  // ---- Phase 1: h = silu(concat @ W_s1^T + b_s1) -> s_x -----------------
  for (int ct = 0; ct < 4; ++ct) {
    int col0 = ct * 16;
    v8f c = gemm_tile_lds(&s_concat[0][0], 193, W_s1, 192, 192, col0, l16, half);
    float bias = b_s1[col0 + l16];
#pragma unroll
    for (int r = 0; r < 8; ++r)
      s_x[r + 8 * half][col0 + l16] = silu(c[r] + bias);
  }
  __syncthreads();

  // ---- Phase 2: dh0 = h @ W_s2^T + b_s2 -> s_concat[:,0:64] + global ----
  for (int ct = 0; ct < 4; ++ct) {
    int col0 = ct * 16;
    v8f c = gemm_tile_lds(&s_x[0][0], 65, W_s2, 64, 64, col0, l16, half);
    float bias = b_s2[col0 + l16];
#pragma unroll
    for (int r = 0; r < 8; ++r) {
      int m = r + 8 * half;
      float x = c[r] + bias;
      s_concat[m][col0 + l16] = x;
      int node = base + m;
      if (node < n_nodes)
        __builtin_nontemporal_store(x, den0 + (size_t)node * 64 + col0 + l16);
    }
  }
  __syncthreads();

  // ---- Phase 3: g = silu(dh0 @ W_g1^T + b_g1) -> s_concat[:,64:128] -----
  for (int ct = 0; ct < 4; ++ct) {
    int col0 = ct * 16;
    v8f c = gemm_tile_lds(&s_concat[0][0], 193, W_g1, 64, 64, col0, l16, half);
    float bias = b_g1[col0 + l16];
#pragma unroll
    for (int r = 0; r < 8; ++r)
      s_concat[r + 8 * half][64 + col0 + l16] = silu(c[r] + bias);
  }
  __syncthreads();

  // ---- Phase 4: scales = g @ W_g2^T + b_g2 -> s_scales ------------------
  for (int ct = 0; ct < 8; ++ct) {
    int col0 = ct * 16;
    v8f c = gemm_tile_lds(&s_concat[0][64], 193, W_g2, 64, 64, col0, l16, half);
    float bias = b_g2[col0 + l16];
#pragma unroll
    for (int r = 0; r < 8; ++r)
      s_scales[r + 8 * half][col0 + l16] = c[r] + bias;
  }
  __syncthreads();

  // ---- Phase 5: delta_h1[:,i,:] = (den1[:,i,:] @ W_L1^T) * alpha1 -------
  for (int i = 0; i < 3; ++i) {
    __syncthreads();
    for (int t = tid; t < 16 * 64; t += 32) {
      int n = t >> 6, f = t & 63;
      int node = base + n;
      if (node >= n_nodes) node = n_nodes - 1;
      s_x[n][f] = den1[(size_t)node * 192 + i * 64 + f];
    }
    __syncthreads();
    for (int ct = 0; ct < 4; ++ct) {
      int col0 = ct * 16;
      v8f c = gemm_tile_lds(&s_x[0][0], 65, W_L1, 64, 64, col0, l16, half);
#pragma unroll
      for (int r = 0; r < 8; ++r) {
        int m = r + 8 * half;
        int node = base + m;
        if (node < n_nodes)
          __builtin_nontemporal_store(
              c[r] * s_scales[m][col0 + l16],
              den1 + (size_t)node * 192 + i * 64 + col0 + l16);
      }
    }
  }

  // ---- Phase 6: delta_h2[:,i,j,:] = (den2[:,ij,:] @ W_L2^T) * alpha2 ----
  for (int ij = 0; ij < 9; ++ij) {
    __syncthreads();
    for (int t = tid; t < 16 * 64; t += 32) {
      int n = t >> 6, f = t & 63;
      int node = base + n;
      if (node >= n_nodes) node = n_nodes - 1;
      s_x[n][f] = den2[(size_t)node * 576 + ij * 64 + f];
    }
    __syncthreads();
    for (int ct = 0; ct < 4; ++ct) {
      int col0 = ct * 16;
      v8f c = gemm_tile_lds(&s_x[0][0], 65, W_L2, 64, 64, col0, l16, half);
#pragma unroll
      for (int r = 0; r < 8; ++r) {
        int m = r + 8 * half;
        int node = base + m;
        if (node < n_nodes)
          __builtin_nontemporal_store(
              c[r] * s_scales[m][64 + col0 + l16],
              den2 + (size_t)node * 576 + ij * 64 + col0 + l16);
      }
    }
  }
}

// ---------------------------------------------------------------------------
// Launcher
// ---------------------------------------------------------------------------
extern "C" void kernel_launch(void* const* d_in, const int* in_sizes, int n_in,
                              void* d_out, int out_size, void* d_ws,
                              size_t ws_size, hipStream_t stream) {
  const float* msg0 = (const float*)d_in[0];
  const float* msg1 = (const float*)d_in[1];
  const float* msg2 = (const float*)d_in[2];
  const int* index = (const int*)d_in[3];
  // d_in[4] = num_nodes (device scalar) -- derived from out_size instead.
  const float* W_s1 = (const float*)d_in[5];
  const float* b_s1 = (const float*)d_in[6];
  const float* W_s2 = (const float*)d_in[7];
  const float* b_s2 = (const float*)d_in[8];
  const float* W_L1 = (const float*)d_in[9];
  const float* W_L2 = (const float*)d_in[10];
  const float* W_g1 = (const float*)d_in[11];
  const float* b_g1 = (const float*)d_in[12];
  const float* W_g2 = (const float*)d_in[13];
  const float* b_g2 = (const float*)d_in[14];
  float* out = (float*)d_out;

  const int n_edges = in_sizes[0] / 64;           // msg0 is (E, 64)
  const int n_nodes = out_size / 832;             // out is N*(64+192+576)

  // Stage A: zero accumulators (out_size is a multiple of 4: 832*N).
  long n4 = (long)out_size / 4;
  zero_f4_kernel<<<(int)((n4 + 255) / 256), 256, 0, stream>>>((float4*)out, n4);

  float* den0 = out;
  float* den1 = out + (size_t)n_nodes * 64;
  float* den2 = out + (size_t)n_nodes * 256;

  // Stage B: scatter-add (L2-resident atomics).
  long t0 = (long)n_edges * 64;
  scatter_add_kernel<64><<<(int)((t0 + 255) / 256), 256, 0, stream>>>(
      msg0, index, den0, t0);
  long t1 = (long)n_edges * 192;
  scatter_add_kernel<192><<<(int)((t1 + 255) / 256), 256, 0, stream>>>(
      msg1, index, den1, t1);
  long t2 = (long)n_edges * 576;
  scatter_add_kernel<576><<<(int)((t2 + 255) / 256), 256, 0, stream>>>(
      msg2, index, den2, t2);

  // Stage C: per-node WMMA math, one wave per 16 nodes, in-place in d_out.
  int nb = (n_nodes + 15) / 16;
  node_update_kernel<<<nb, 32, 0, stream>>>(out, n_nodes, W_s1, b_s1, W_s2,
                                            b_s2, W_g1, b_g1, W_g2, b_g2, W_L1,
                                            W_L2);
}